// DAriEL_Decoder_Cell_1_88064009437441
// MI455X (gfx1250) — compile-verified
//
#include <hip/hip_runtime.h>
#include <hip/hip_bf16.h>
#include <math.h>

#define BATCH    16
#define EMBV     256
#define UNITS    2048
#define KTOT     2304      // EMB + UNITS
#define NTOT     8192      // 4*UNITS
#define MAXLEN   8
#define KC       288       // K chunk staged in LDS (16*288*4 = 18 KB)
#define KSEG     4         // deterministic split-K factor
#define KSEG_LEN (KTOT / KSEG)   // 576 = 2 * KC

typedef __attribute__((ext_vector_type(2))) float v2f;
typedef __attribute__((ext_vector_type(8))) float v8f;

__device__ __forceinline__ float sigm(float x) { return 1.0f / (1.0f + expf(-x)); }

// ---------------------------------------------------------------------------
// Tokens: uniform initial softmax -> literal cumsum interval search (exact,
// since 1/2048 = 2^-11 and all partial sums are exactly representable).
// token[b][t] uses v = input_point[b][t]  (curDim == t because MAXLEN < LAT)
// ---------------------------------------------------------------------------
__global__ void token_kernel(const float* __restrict__ input_point,
                             int* __restrict__ tokens) {
    int tid = threadIdx.x;
    if (tid >= BATCH * MAXLEN) return;
    int b = tid >> 3, t = tid & 7;
    float v = input_point[b * 16 + t];
    const float p = 1.0f / 2048.0f;
    float clo = 0.0f;
    int tok = 0;                      // argmax of all-False == 0
    for (int k = 0; k < 2048; ++k) {
        float chi = clo + p;
        if (clo <= v && v <= chi) { tok = k; break; }
        clo = chi;
    }
    tokens[tid] = tok;
}

// ---------------------------------------------------------------------------
// Partial Z_s = X[:, ks:ks+576] @ [Wk;Wr][ks:ks+576, :]  via
// V_WMMA_F32_16X16X4_F32.  X = [E[token_t] | h_prev] is gathered on the fly
// while staging into LDS (no X buffer).  Grid = (32, KSEG); block = 8 waves,
// each wave owns 2 consecutive 16-col N tiles.
// ---------------------------------------------------------------------------
__global__ __launch_bounds__(256) void gemm_gates_kernel(
        const float* __restrict__ E,
        const float* __restrict__ h,
        const int*   __restrict__ tokens, int t,
        const float* __restrict__ Wk,
        const float* __restrict__ Wr,
        float* __restrict__ Zp) {
    __shared__ __attribute__((aligned(16))) float sX[16 * KC];

    const int lane  = threadIdx.x & 31;
    const int wave  = threadIdx.x >> 5;
    const int hl    = lane >> 4;      // lane half: 0 or 1
    const int l16   = lane & 15;
    const int nbase = blockIdx.x * 256 + wave * 32;    // 2 tiles of 16 cols
    const int kstart = blockIdx.y * KSEG_LEN;

    v8f acc0 = {};
    v8f acc1 = {};

    for (int kc = kstart; kc < kstart + KSEG_LEN; kc += KC) {
        // stage X[:, kc:kc+KC] into LDS, gathering from E[token] / h directly
        for (int idx = threadIdx.x; idx < 16 * KC; idx += 256) {
            int row = idx / KC, cc = idx - row * KC;
            int j = kc + cc;
            float v;
            if (j < EMBV) {
                int tok = tokens[row * MAXLEN + t];
                v = E[tok * EMBV + j];
            } else {
                v = h[row * UNITS + (j - EMBV)];
            }
            sX[idx] = v;
        }
        __syncthreads();

        for (int k4 = 0; k4 < KC; k4 += 4) {
            const int kk = kc + k4;
            // Wk/Wr split at row 256 is 4-aligned -> wave-uniform select
            const float* Wp; int rb;
            if (kk < EMBV) { Wp = Wk; rb = kk; }
            else           { Wp = Wr; rb = kk - EMBV; }
            const int wr = rb + 2 * hl;

            // A fragment: lanes 0-15 hold K=k4,k4+1 ; lanes 16-31 K=k4+2,k4+3
            const float2 av =
                *reinterpret_cast<const float2*>(&sX[l16 * KC + k4 + 2 * hl]);
            v2f a; a.x = av.x; a.y = av.y;

            // B fragments: half-wave hl supplies rows wr, wr+1 at col nbase+l16
            const float* bp = Wp + (size_t)wr * NTOT + nbase + l16;
            v2f b0, b1;
            b0.x = bp[0];   b0.y = bp[NTOT];
            b1.x = bp[16];  b1.y = bp[NTOT + 16];

            acc0 = __builtin_amdgcn_wmma_f32_16x16x4_f32(
                       false, a, false, b0, (short)0, acc0, false, false);
            acc1 = __builtin_amdgcn_wmma_f32_16x16x4_f32(
                       false, a, false, b1, (short)0, acc1, false, false);
        }
        __syncthreads();
    }

    // C/D layout: VGPR r -> (M=r, N=lane) lanes 0-15 ; (M=r+8, N=lane-16)
    float* Zpart = Zp + (size_t)blockIdx.y * BATCH * NTOT;
#pragma unroll
    for (int r = 0; r < 8; ++r) {
        int m = r + 8 * hl;
        Zpart[m * NTOT + nbase + l16]      = acc0[r];
        Zpart[m * NTOT + nbase + 16 + l16] = acc1[r];
    }
}

// ---------------------------------------------------------------------------
// Fused: fixed-order split-K reduction + bias + LSTM gates + PAD mask +
// state store + row softmax into out[b, t, :].  One block per batch row;
// each thread owns 8 units in registers.
// ---------------------------------------------------------------------------
__global__ __launch_bounds__(256) void lstm_softmax_kernel(
        const float* __restrict__ Zp,
        const float* __restrict__ bias,
        const int*   __restrict__ tokens, int t,
        float* __restrict__ h,
        float* __restrict__ c,
        float* __restrict__ out) {
    __shared__ float red[256];
    const int b   = blockIdx.x;
    const int tok = tokens[b * MAXLEN + t];

    float hv[8];
    float mx = -INFINITY;
#pragma unroll
    for (int k = 0; k < 8; ++k) {
        const int u = threadIdx.x + k * 256;
        const size_t base = (size_t)b * NTOT + u;

        float zi = 0.f, zf = 0.f, zg = 0.f, zo = 0.f;
#pragma unroll
        for (int s = 0; s < KSEG; ++s) {   // fixed order -> deterministic
            const float* zr = Zp + (size_t)s * BATCH * NTOT + base;
            zi += zr[0];
            zf += zr[2048];
            zg += zr[4096];
            zo += zr[6144];
        }
        zi += bias[u];
        zf += bias[u + 2048];
        zg += bias[u + 4096];
        zo += bias[u + 6144];

        const int idx = b * UNITS + u;
        float cold = c[idx];
        float cn = sigm(zf) * cold + sigm(zi) * tanhf(zg);
        float hn = sigm(zo) * tanhf(cn);
        float cv  = (tok != 0) ? cn : cold;     // mask_zero: PAD keeps state
        float hvk = (tok != 0) ? hn : h[idx];
        c[idx] = cv;
        h[idx] = hvk;
        hv[k] = hvk;
        mx = fmaxf(mx, hvk);
    }

    // block max over the 2048-wide row
    red[threadIdx.x] = mx; __syncthreads();
    for (int s = 128; s > 0; s >>= 1) {
        if (threadIdx.x < s)
            red[threadIdx.x] = fmaxf(red[threadIdx.x], red[threadIdx.x + s]);
        __syncthreads();
    }
    mx = red[0]; __syncthreads();

    float e[8];
    float sum = 0.f;
#pragma unroll
    for (int k = 0; k < 8; ++k) { e[k] = expf(hv[k] - mx); sum += e[k]; }

    red[threadIdx.x] = sum; __syncthreads();
    for (int s = 128; s > 0; s >>= 1) {
        if (threadIdx.x < s)
            red[threadIdx.x] += red[threadIdx.x + s];
        __syncthreads();
    }
    sum = red[0];

    float* orow = out + ((size_t)b * MAXLEN + t) * UNITS;
#pragma unroll
    for (int k = 0; k < 8; ++k)
        orow[threadIdx.x + k * 256] = e[k] / sum;
}

// ---------------------------------------------------------------------------
extern "C" void kernel_launch(void* const* d_in, const int* in_sizes, int n_in,
                              void* d_out, int out_size, void* d_ws, size_t ws_size,
                              hipStream_t stream) {
    const float* input_point = (const float*)d_in[0];   // (16,16)
    const float* E           = (const float*)d_in[1];   // (2048,256)
    const float* Wk          = (const float*)d_in[2];   // (256,8192)
    const float* Wr          = (const float*)d_in[3];   // (2048,8192)
    const float* bias        = (const float*)d_in[4];   // (8192,)
    float* out = (float*)d_out;                         // (16,8,2048)

    char* ws = (char*)d_ws;
    int*   tokens = (int*)ws;                                   //   512 B
    float* h      = (float*)(ws + 512);                         // 128 KB
    float* c      = (float*)(ws + 512 + 131072);                // 128 KB
    float* Zp     = (float*)(ws + 512 + 2 * 131072);            // KSEG * 512 KB

    // zero LSTM state each call (h and c are contiguous)
    hipMemsetAsync(h, 0, 2 * 131072, stream);

    token_kernel<<<1, 128, 0, stream>>>(input_point, tokens);

    for (int t = 0; t < MAXLEN; ++t) {
        gemm_gates_kernel<<<dim3(NTOT / 256, KSEG), 256, 0, stream>>>(
            E, h, tokens, t, Wk, Wr, Zp);
        lstm_softmax_kernel<<<BATCH, 256, 0, stream>>>(
            Zp, bias, tokens, t, h, c, out);
    }
}